// GraphConv_71614284694203
// MI455X (gfx1250) — compile-verified
//
#include <hip/hip_runtime.h>

// GraphConv: out[b,n,f] = sum_{k,d} x[b, nbr[n,k], d] * W[k,d,f] + bias[f]
// B=4, N=50000, K=16, D=32, F=32.  Cast as tall-skinny GEMM (M=B*N, Kdim=512, N=32)
// on fp32 WMMA 16x16x4.  Gather loads land directly in the WMMA A-fragment layout.

typedef __attribute__((ext_vector_type(2))) float v2f;
typedef __attribute__((ext_vector_type(8))) float v8f;

#define BB 4
#define NN 50000
#define KK 16
#define DD 32
#define FF 32
#define KDIM (KK * DD)          // 512
#define NGROUP (KDIM / 4)       // 128 kd-groups of 4
#define TILES_PER_BATCH (NN / 16)  // 3125
#define TOTAL_TILES (BB * TILES_PER_BATCH)  // 12500
#define WAVES_PER_BLOCK 8

__global__ __launch_bounds__(256)
void GraphConv_wmma_f32(const float* __restrict__ x,
                        const int*   __restrict__ nbr,
                        const float* __restrict__ W,
                        const float* __restrict__ bias,
                        float*       __restrict__ out) {
    // B fragments pre-swizzled: frag[g][tile][lane] = { W[4g+2*hi+0][f], W[4g+2*hi+1][f] }
    // with hi = lane>>4, f = (lane&15) + 16*tile.  128*2*32 float2 = 64 KB.
    __shared__ v2f Wfrag[NGROUP * 2 * 32];

    for (int i = threadIdx.x; i < NGROUP * 2 * 32; i += blockDim.x) {
        int lane_i = i & 31;
        int tile   = (i >> 5) & 1;
        int g      = i >> 6;
        int f      = (lane_i & 15) + tile * 16;
        int krow   = 4 * g + ((lane_i >> 4) << 1);
        v2f wv;
        wv.x = W[(krow + 0) * FF + f];
        wv.y = W[(krow + 1) * FF + f];
        Wfrag[i] = wv;
    }
    __syncthreads();

    const int wave  = threadIdx.x >> 5;
    const int lane  = threadIdx.x & 31;
    const int gtile = blockIdx.x * WAVES_PER_BLOCK + wave;
    if (gtile >= TOTAL_TILES) return;   // whole-wave exit; active waves keep EXEC all-1s

    const int b         = gtile / TILES_PER_BATCH;
    const int node_base = (gtile % TILES_PER_BATCH) * 16;
    const int m         = lane & 15;    // matrix row (node within tile) / output column
    const int hi        = lane >> 4;    // selects K-subcolumns {2,3} vs {0,1}
    const int node      = node_base + m;

    // 16 neighbor indices of this lane's node (lanes l and l+16 duplicate — fine)
    int nb[KK];
    {
        const int4* np = (const int4*)(nbr + node * KK);
        #pragma unroll
        for (int q = 0; q < 4; ++q) {
            int4 v = np[q];
            nb[4*q+0] = v.x; nb[4*q+1] = v.y; nb[4*q+2] = v.z; nb[4*q+3] = v.w;
        }
    }

    v8f acc0 = {};  // F columns 0..15
    v8f acc1 = {};  // F columns 16..31

    const float* xb = x + (size_t)b * NN * DD;

    for (int k = 0; k < KK; ++k) {
        // Row of x for this lane's node's k-th neighbor, offset so that a single
        // b64 load yields A-fragment elements (K = 4t + 2*hi + {0,1}).
        const float* row = xb + (size_t)nb[k] * DD + hi * 2;
        #pragma unroll
        for (int t = 0; t < 8; ++t) {
            const int g = k * 8 + t;               // kd-group, kd = 4g
            v2f a  = *(const v2f*)(row + t * 4);   // global_load_b64 gather -> A fragment
            v2f b0 = Wfrag[(g * 2 + 0) * 32 + lane];  // ds_load_b64, conflict-free
            v2f b1 = Wfrag[(g * 2 + 1) * 32 + lane];
            acc0 = __builtin_amdgcn_wmma_f32_16x16x4_f32(
                       false, a, false, b0, (short)0, acc0, false, false);
            acc1 = __builtin_amdgcn_wmma_f32_16x16x4_f32(
                       false, a, false, b1, (short)0, acc1, false, false);
        }
    }

    // C/D layout: lane holds column N=m, rows M = v + 8*hi for VGPR v.
    const float bias0 = bias[m];
    const float bias1 = bias[m + 16];
    float* ob = out + ((size_t)b * NN + node_base) * FF;
    #pragma unroll
    for (int v = 0; v < 8; ++v) {
        const int row_m = v + hi * 8;
        ob[row_m * FF + m]      = acc0[v] + bias0;
        ob[row_m * FF + m + 16] = acc1[v] + bias1;
    }
}

extern "C" void kernel_launch(void* const* d_in, const int* in_sizes, int n_in,
                              void* d_out, int out_size, void* d_ws, size_t ws_size,
                              hipStream_t stream) {
    const float* x    = (const float*)d_in[0];
    const int*   nbr  = (const int*)  d_in[1];
    const float* W    = (const float*)d_in[2];
    const float* bias = (const float*)d_in[3];
    float*       out  = (float*)d_out;

    const int blocks = (TOTAL_TILES + WAVES_PER_BLOCK - 1) / WAVES_PER_BLOCK; // 1563
    GraphConv_wmma_f32<<<blocks, 256, 0, stream>>>(x, nbr, W, bias, out);
}